// SimpleHGNConv_36429912605266
// MI455X (gfx1250) — compile-verified
//
#include <hip/hip_runtime.h>
#include <math.h>

// ---------------------------------------------------------------------------
// SimpleHGN conv for MI455X (gfx1250, wave32)
// Dims fixed by reference: N=50000, E=800000, IN_DIM=128, H=4, D=32,
// EDGE_DIM=32, T=8, D*H = 128.
// ---------------------------------------------------------------------------

#define IN_DIM   128
#define NHEAD    4
#define DHEAD    32
#define FEAT     128      // D*H
#define EDGE_DIM 32
#define NTYPE    8
#define NEG_SLOPE 0.2f

typedef __attribute__((ext_vector_type(2))) float v2f;
typedef __attribute__((ext_vector_type(8))) float v8f;

// ---------------------------------------------------------------------------
// Kernel 1: dual WMMA GEMM.  emb = h @ W ; res = h @ res_w + res_b
// One wave per 16x16 output tile; 8 waves/block cover the 8 column tiles of a
// 16-row strip.  f32 WMMA 16x16x4, K looped 0..124 step 4 (32 wmma per acc).
// ---------------------------------------------------------------------------
__global__ __launch_bounds__(256) void hgn_gemm_proj(
    const float* __restrict__ h,    // [N,128]
    const float* __restrict__ W,    // [128,128]
    const float* __restrict__ Rw,   // [128,128]
    const float* __restrict__ Rb,   // [128]
    float* __restrict__ emb,        // [N,128]
    float* __restrict__ res,        // [N,128]
    int N)
{
    const int wave  = threadIdx.x >> 5;        // 0..7 -> column tile
    const int lane  = threadIdx.x & 31;
    const int row0  = blockIdx.x * 16;
    const int col0  = wave * 16;
    const int lhalf = lane >> 4;               // 0: lanes 0-15, 1: lanes 16-31
    const int l     = lane & 15;

    // A-row this lane feeds (clamp for safety; N is a multiple of 16 here)
    int arow = row0 + l; if (arow >= N) arow = N - 1;
    const float* __restrict__ hrow = h + (size_t)arow * IN_DIM;

    v8f accW = {};
    v8f accR = {};

    #pragma unroll 4
    for (int k = 0; k < IN_DIM; k += 4) {
        const int kk = k + 2 * lhalf;
        // A 16x4: lanes 0-15 hold K=k..k+1, lanes 16-31 hold K=k+2..k+3
        v2f a;
        a.x = hrow[kk + 0];
        a.y = hrow[kk + 1];
        // B 4x16 mirrored: this lane supplies rows kk,kk+1 at column col0+l
        const float* __restrict__ wp = W  + (size_t)kk * FEAT + (col0 + l);
        const float* __restrict__ rp = Rw + (size_t)kk * FEAT + (col0 + l);
        v2f bW; bW.x = wp[0]; bW.y = wp[FEAT];
        v2f bR; bR.x = rp[0]; bR.y = rp[FEAT];
        // 8 args: (neg_a, A, neg_b, B, c_mod, C, reuse_a, reuse_b)
        accW = __builtin_amdgcn_wmma_f32_16x16x4_f32(
                   false, a, false, bW, (short)0, accW, false, false);
        accR = __builtin_amdgcn_wmma_f32_16x16x4_f32(
                   false, a, false, bR, (short)0, accR, false, false);
    }

    const float bias = Rb[col0 + l];
    // C/D layout: VGPR v -> row M = v + 8*lhalf, col N = l
    if (row0 + 16 <= N) {
        // fast path: uniform, no per-row guards -> straight-line stores
        #pragma unroll
        for (int v = 0; v < 8; ++v) {
            const int m = v + 8 * lhalf;
            const size_t idx = (size_t)(row0 + m) * FEAT + (col0 + l);
            emb[idx] = accW[v];
            res[idx] = accR[v] + bias;
        }
    } else {
        #pragma unroll
        for (int v = 0; v < 8; ++v) {
            const int m = v + 8 * lhalf;
            const int r = row0 + m;
            if (r < N) {
                const size_t idx = (size_t)r * FEAT + (col0 + l);
                emb[idx] = accW[v];
                res[idx] = accR[v] + bias;
            }
        }
    }
}

// ---------------------------------------------------------------------------
// Kernel 2: per-edge-type features.  e_t[t,f] = sum_e edge_emb[t,e]*W_r[t,e,f]
//           h_e_t[t,h] = sum_ed a_e[h,ed] * e_t[t, h*32+ed]
// Tiny (8*128 outputs): single block.
// ---------------------------------------------------------------------------
__global__ __launch_bounds__(256) void hgn_edge_type(
    const float* __restrict__ edge_emb,  // [T,32]
    const float* __restrict__ W_r,       // [T,32,128]
    const float* __restrict__ a_e,       // [H,32]
    float* __restrict__ h_e_t)           // [T,H]
{
    __shared__ float et[NTYPE][FEAT];
    const int tid = threadIdx.x;
    for (int idx = tid; idx < NTYPE * FEAT; idx += blockDim.x) {
        const int t = idx >> 7, f = idx & 127;
        float s = 0.f;
        #pragma unroll 8
        for (int e = 0; e < EDGE_DIM; ++e)
            s += edge_emb[t * EDGE_DIM + e] * W_r[((size_t)t * EDGE_DIM + e) * FEAT + f];
        et[t][f] = s;
    }
    __syncthreads();
    if (tid < NTYPE * NHEAD) {
        const int t = tid >> 2, hh = tid & 3;
        float s = 0.f;
        #pragma unroll 8
        for (int e = 0; e < EDGE_DIM; ++e)
            s += a_e[hh * EDGE_DIM + e] * et[t][hh * EDGE_DIM + e];
        h_e_t[t * NHEAD + hh] = s;
    }
}

// ---------------------------------------------------------------------------
// Kernel 3: per-node attention dots + init of softmax/agg state.
// One thread per (n, head).
// ---------------------------------------------------------------------------
__global__ __launch_bounds__(256) void hgn_node_init(
    const float* __restrict__ emb,   // [N,128]
    const float* __restrict__ a_l,   // [H,32]
    const float* __restrict__ a_r,   // [H,32]
    float* __restrict__ h_l,         // [N,H]
    float* __restrict__ h_r,         // [N,H]
    float* __restrict__ m,           // [N,H]
    float* __restrict__ denom,       // [N,H]
    float* __restrict__ agg,         // [N,H,D] zeroed here
    int N)
{
    const int i = blockIdx.x * blockDim.x + threadIdx.x;
    if (i >= N * NHEAD) return;
    const int n = i >> 2, hh = i & 3;
    const float* __restrict__ ep = emb + (size_t)n * FEAT + hh * DHEAD;
    float sl = 0.f, sr = 0.f;
    #pragma unroll 8
    for (int d = 0; d < DHEAD; ++d) {
        const float v = ep[d];
        sl += a_l[hh * DHEAD + d] * v;
        sr += a_r[hh * DHEAD + d] * v;
    }
    h_l[i] = sl;
    h_r[i] = sr;
    m[i] = __int_as_float(0xFF800000);   // -inf
    denom[i] = 0.f;
    float* __restrict__ ap = agg + (size_t)n * FEAT + hh * DHEAD;
    #pragma unroll 8
    for (int d = 0; d < DHEAD; ++d) ap[d] = 0.f;
}

// Float atomic-max via integer ordering trick (works for all signs).
__device__ __forceinline__ void atomicMaxF32(float* addr, float val) {
    if (val >= 0.f) atomicMax((int*)addr, __float_as_int(val));
    else            atomicMin((unsigned int*)addr, __float_as_uint(val));
}

// ---------------------------------------------------------------------------
// Kernel 4: edge scores + leaky relu + atomic segment max over destination.
// One thread per (edge, head).
// ---------------------------------------------------------------------------
__global__ __launch_bounds__(256) void hgn_edge_score(
    const float* __restrict__ h_l, const float* __restrict__ h_r,
    const float* __restrict__ h_e_t,
    const int* __restrict__ row, const int* __restrict__ col,
    const int* __restrict__ etype,
    float* __restrict__ sc,          // [E,H]
    float* __restrict__ m,           // [N,H]
    int E)
{
    const int i = blockIdx.x * blockDim.x + threadIdx.x;
    if (i >= E * NHEAD) return;
    const int e = i >> 2, hh = i & 3;
    const int r = row[e], c = col[e], t = etype[e];
    float s = h_l[r * NHEAD + hh] + h_r[c * NHEAD + hh] + h_e_t[t * NHEAD + hh];
    s = (s > 0.f) ? s : NEG_SLOPE * s;
    sc[i] = s;
    atomicMaxF32(&m[c * NHEAD + hh], s);
}

// ---------------------------------------------------------------------------
// Kernel 5: ex = exp(s - m[col]); atomic denom accumulate.
// ---------------------------------------------------------------------------
__global__ __launch_bounds__(256) void hgn_edge_exp(
    const int* __restrict__ col,
    const float* __restrict__ m,
    float* __restrict__ sc,          // in: scores, out: exp
    float* __restrict__ denom,
    int E)
{
    const int i = blockIdx.x * blockDim.x + threadIdx.x;
    if (i >= E * NHEAD) return;
    const int e = i >> 2, hh = i & 3;
    const int c = col[e];
    const float ex = __expf(sc[i] - m[c * NHEAD + hh]);
    sc[i] = ex;
    atomicAdd(&denom[c * NHEAD + hh], ex);
}

// ---------------------------------------------------------------------------
// Kernel 6: message scatter.  One wave (32 lanes) per edge; each lane handles
// one feature per head iteration -> fully coalesced 128B gather from emb[row]
// and 128B of global_atomic_add_f32 into agg[col].  emb/agg live in L2.
// ---------------------------------------------------------------------------
__global__ __launch_bounds__(256) void hgn_edge_msg(
    const float* __restrict__ emb,    // [N,128]
    const float* __restrict__ sc,     // [E,H] (exp values)
    const float* __restrict__ denom,  // [N,H]
    const int* __restrict__ row, const int* __restrict__ col,
    float* __restrict__ agg,          // [N,H,D]
    int E)
{
    const int wave = (blockIdx.x * blockDim.x + threadIdx.x) >> 5;
    const int lane = threadIdx.x & 31;
    if (wave >= E) return;
    const int r = row[wave], c = col[wave];
    const float* __restrict__ ep = emb + (size_t)r * FEAT;
    float* __restrict__ ap = agg + (size_t)c * FEAT;
    #pragma unroll
    for (int hh = 0; hh < NHEAD; ++hh) {
        const float alpha = sc[(size_t)wave * NHEAD + hh] / denom[c * NHEAD + hh];
        const int f = hh * DHEAD + lane;
        atomicAdd(&ap[f], ep[f] * alpha);
    }
}

// ---------------------------------------------------------------------------
// Kernel 7: finalize.  out[n, d*H+h] = elu(agg[n,h,d] + res[n, d*H+h])
// ---------------------------------------------------------------------------
__global__ __launch_bounds__(256) void hgn_finalize(
    const float* __restrict__ agg,   // [N,H,D]
    const float* __restrict__ res,   // [N,128] (bias already folded)
    float* __restrict__ out,         // [N,128] d-major
    int N)
{
    const int i = blockIdx.x * blockDim.x + threadIdx.x;
    if (i >= N * FEAT) return;
    const int n = i >> 7, j = i & 127;
    const int hh = j & 3, d = j >> 2;           // j = d*H + h
    const float x = agg[(size_t)n * FEAT + hh * DHEAD + d] + res[i];
    out[i] = (x > 0.f) ? x : (__expf(x) - 1.f);
}

// ---------------------------------------------------------------------------
// Launch
// ---------------------------------------------------------------------------
extern "C" void kernel_launch(void* const* d_in, const int* in_sizes, int n_in,
                              void* d_out, int out_size, void* d_ws, size_t ws_size,
                              hipStream_t stream)
{
    const float* h        = (const float*)d_in[0];   // [N,128]
    const float* W        = (const float*)d_in[1];   // [128,128]
    const float* edge_emb = (const float*)d_in[2];   // [8,32]
    const float* W_r      = (const float*)d_in[3];   // [8,32,128]
    const float* a_l      = (const float*)d_in[4];   // [1,4,32]
    const float* a_r      = (const float*)d_in[5];   // [1,4,32]
    const float* a_e      = (const float*)d_in[6];   // [1,4,32]
    const float* res_w    = (const float*)d_in[7];   // [128,128]
    const float* res_b    = (const float*)d_in[8];   // [128]
    const int*   row      = (const int*)d_in[9];     // [E]
    const int*   col      = (const int*)d_in[10];    // [E]
    const int*   etype    = (const int*)d_in[11];    // [E]

    const int N = in_sizes[0] / IN_DIM;
    const int E = in_sizes[9];

    // workspace carve-up (floats)
    float* ws    = (float*)d_ws;
    float* emb   = ws;                         // N*128
    float* res   = emb   + (size_t)N * FEAT;   // N*128
    float* agg   = res   + (size_t)N * FEAT;   // N*128
    float* h_l   = agg   + (size_t)N * FEAT;   // N*4
    float* h_r   = h_l   + (size_t)N * NHEAD;  // N*4
    float* m     = h_r   + (size_t)N * NHEAD;  // N*4
    float* denom = m     + (size_t)N * NHEAD;  // N*4
    float* sc    = denom + (size_t)N * NHEAD;  // E*4
    float* h_e_t = sc    + (size_t)E * NHEAD;  // T*H

    // 1) dual WMMA GEMM (emb, res)
    {
        const int rowTiles = (N + 15) / 16;
        hgn_gemm_proj<<<rowTiles, 256, 0, stream>>>(h, W, res_w, res_b, emb, res, N);
    }
    // 2) per-type edge features
    hgn_edge_type<<<1, 256, 0, stream>>>(edge_emb, W_r, a_e, h_e_t);
    // 3) node dots + state init
    {
        const int total = N * NHEAD;
        hgn_node_init<<<(total + 255) / 256, 256, 0, stream>>>(
            emb, a_l, a_r, h_l, h_r, m, denom, agg, N);
    }
    // 4) scores + segment max
    {
        const int total = E * NHEAD;
        hgn_edge_score<<<(total + 255) / 256, 256, 0, stream>>>(
            h_l, h_r, h_e_t, row, col, etype, sc, m, E);
    }
    // 5) exp + segment sum
    {
        const int total = E * NHEAD;
        hgn_edge_exp<<<(total + 255) / 256, 256, 0, stream>>>(col, m, sc, denom, E);
    }
    // 6) message scatter (one wave per edge)
    {
        const int wavesPerBlock = 8;                  // 256 threads
        const int blocks = (E + wavesPerBlock - 1) / wavesPerBlock;
        hgn_edge_msg<<<blocks, 256, 0, stream>>>(emb, sc, denom, row, col, agg, E);
    }
    // 7) transpose + residual + ELU
    {
        const int total = N * FEAT;
        hgn_finalize<<<(total + 255) / 256, 256, 0, stream>>>(agg, res, (float*)d_out, N);
    }
}